// GraphConv_73933567034039
// MI455X (gfx1250) — compile-verified
//
#include <hip/hip_runtime.h>
#include <hip/hip_bf16.h>

typedef __attribute__((ext_vector_type(2))) float v2f;
typedef __attribute__((ext_vector_type(8))) float v8f;

#define DIM 128

// -------------------------------------------------------------------------
// Kernel 1: h = x @ W  via V_WMMA_F32_16X16X4_F32 (exact f32 matrix math)
// 256 threads = 8 waves; each wave computes a 16x128 strip of h.
// W is staged transposed in LDS (wT[n*128+k]) so B fragments are ds_load_b64.
// -------------------------------------------------------------------------
__global__ __launch_bounds__(256) void gcn_gemm_wmma(
    const float* __restrict__ x, const float* __restrict__ w,
    float* __restrict__ h, int n_nodes)
{
    __shared__ float wT[DIM * DIM];   // 64 KB (WGP has 320 KB)

    const int tid = threadIdx.x;
    // Stage W transposed: wT[n*128 + k] = w[k*128 + n]  (coalesced global read)
    for (int i = tid; i < DIM * DIM; i += 256) {
        int k = i >> 7;
        int n = i & 127;
        wT[n * DIM + k] = w[i];
    }
    __syncthreads();

    const int wave = tid >> 5;
    const int lane = tid & 31;
    const int m    = lane & 15;    // row-in-tile (A) / col-in-tile (B,D)
    const int kh   = lane >> 4;    // K half: 0 -> K=0,1 ; 1 -> K=2,3

    const int row0 = (blockIdx.x * 8 + wave) * 16;
    int rowm = row0 + m;
    if (rowm >= n_nodes) rowm = n_nodes - 1;   // clamp: keep EXEC all-1s for WMMA

    const float* xrow = x + (size_t)rowm * DIM;

    v8f acc[8];
#pragma unroll
    for (int ct = 0; ct < 8; ++ct) acc[ct] = (v8f)(0.0f);

#pragma unroll 4
    for (int kk = 0; kk < DIM; kk += 4) {
        // A fragment 16x4: lane m, half kh -> x[row][kk+2kh], x[row][kk+2kh+1]
        v2f a;
        a.x = xrow[kk + 2 * kh + 0];
        a.y = xrow[kk + 2 * kh + 1];
#pragma unroll
        for (int ct = 0; ct < 8; ++ct) {
            // B fragment 4x16: col ct*16+m, rows kk+2kh, kk+2kh+1 (contiguous in wT)
            const float* bp = &wT[(ct * 16 + m) * DIM + kk + 2 * kh];
            v2f b;
            b.x = bp[0];
            b.y = bp[1];
            acc[ct] = __builtin_amdgcn_wmma_f32_16x16x4_f32(
                false, a, false, b, (short)0, acc[ct], false, false);
        }
    }

    // D layout: VGPR i -> row = row0 + 8*kh + i, col = ct*16 + m
    const int mOut = 8 * kh;
#pragma unroll
    for (int ct = 0; ct < 8; ++ct) {
#pragma unroll
        for (int i = 0; i < 8; ++i) {
            int r = row0 + mOut + i;
            if (r < n_nodes)
                h[(size_t)r * DIM + ct * 16 + m] = acc[ct][i];
        }
    }
}

// -------------------------------------------------------------------------
// Kernel 2: out[i][:] = bias[:]   (also re-initializes the poisoned d_out)
// -------------------------------------------------------------------------
__global__ __launch_bounds__(256) void gcn_bias_init(
    const float* __restrict__ bias, float* __restrict__ out, long long total)
{
    long long i = (long long)blockIdx.x * 256 + threadIdx.x;
    if (i < total) out[i] = bias[i & (DIM - 1)];
}

// -------------------------------------------------------------------------
// Kernel 3: edge scatter: out[row[e]] += edge_vals[e] * h[col[e]]
// One wave per edge: lanes cooperatively read the 512B source row as float4
// (coalesced global_load_b128) and issue no-return global_atomic_add_f32.
// h (51.2 MB) and out (51.2 MB) both live in the 192 MB L2.
// -------------------------------------------------------------------------
__global__ __launch_bounds__(256) void gcn_edge_scatter(
    const float* __restrict__ h, const float* __restrict__ ev,
    const int* __restrict__ row, const int* __restrict__ col,
    float* out, int n_edges)
{
    int e = blockIdx.x * 8 + (threadIdx.x >> 5);
    if (e >= n_edges) return;
    const int lane = threadIdx.x & 31;

    const float  w = ev[e];
    const size_t src = (size_t)col[e] * DIM;
    const size_t dst = (size_t)row[e] * DIM;

    const float4 v = *((const float4*)(h + src) + lane);
    float* o = out + dst + lane * 4;
    atomicAdd(o + 0, w * v.x);
    atomicAdd(o + 1, w * v.y);
    atomicAdd(o + 2, w * v.z);
    atomicAdd(o + 3, w * v.w);
}

// -------------------------------------------------------------------------
extern "C" void kernel_launch(void* const* d_in, const int* in_sizes, int n_in,
                              void* d_out, int out_size, void* d_ws, size_t ws_size,
                              hipStream_t stream)
{
    const float* x    = (const float*)d_in[0];   // [N, 128]
    const float* w    = (const float*)d_in[1];   // [128, 128]
    const float* bias = (const float*)d_in[2];   // [1, 128]
    const float* ev   = (const float*)d_in[3];   // [E]
    const int*   row  = (const int*)d_in[4];     // [E]
    const int*   col  = (const int*)d_in[5];     // [E]

    const int n_nodes = in_sizes[0] / DIM;
    const int n_edges = in_sizes[3];

    float* h   = (float*)d_ws;                   // [N, 128] scratch
    float* out = (float*)d_out;

    // 1) h = x @ W  (WMMA f32)
    int gemm_blocks = (n_nodes + 127) / 128;     // 8 waves * 16 rows per block
    gcn_gemm_wmma<<<gemm_blocks, 256, 0, stream>>>(x, w, h, n_nodes);

    // 2) out = broadcast(bias)
    long long total = (long long)n_nodes * DIM;
    int bias_blocks = (int)((total + 255) / 256);
    gcn_bias_init<<<bias_blocks, 256, 0, stream>>>(bias, out, total);

    // 3) out[row[e]] += ev[e] * h[col[e]]
    int sc_blocks = (n_edges + 7) / 8;           // one wave per edge
    gcn_edge_scatter<<<sc_blocks, 256, 0, stream>>>(h, ev, row, col, out, n_edges);
}